// GNN_LSTM_36807869727434
// MI455X (gfx1250) — compile-verified
//
#include <hip/hip_runtime.h>
#include <math.h>
#include <stdint.h>

// Problem constants (match reference)
#define BB    16
#define NNODE 400
#define FDIM  400
#define HD    128
#define TT    8
#define TSL   200
#define EDEG  32
#define NTOT  (BB*NNODE)     // 6400
#define ETOT  (NTOT*EDEG)    // 204800
#define KTOP  200
#define KCHUNK 64
#define LDP   65             // padded LDS stride (conflict-free ds_load_b64)

typedef __attribute__((ext_vector_type(2))) float v2f;
typedef __attribute__((ext_vector_type(8))) float v8f;

// Async global->LDS staging if the toolchain declares the gfx1250 builtins.
#if defined(__HIP_DEVICE_COMPILE__) && \
    __has_builtin(__builtin_amdgcn_global_load_async_to_lds_b32) && \
    __has_builtin(__builtin_amdgcn_s_wait_asynccnt)
#define USE_ASYNC_LDS 1
#else
#define USE_ASYNC_LDS 0
#endif

#if USE_ASYNC_LDS
typedef __attribute__((address_space(1))) float gfloat_t;  // global (AS1)
typedef __attribute__((address_space(3))) float lfloat_t;  // LDS (AS3)
typedef __attribute__((address_space(1))) int   gint_t;
typedef __attribute__((address_space(3))) int   lint_t;
__device__ __forceinline__ void async_ld_f32(const float* g, float* l){
  gfloat_t* gf = (gfloat_t*)g;   // addrspacecast (generic -> AS1)
  lfloat_t* lf = (lfloat_t*)l;   // addrspacecast (generic -> AS3)
  __builtin_amdgcn_global_load_async_to_lds_b32((gint_t*)gf, (lint_t*)lf, 0, 0);
}
#endif

struct Bias8 { const float* p[8]; };

__device__ __forceinline__ float sigmf(float x){ return 1.0f/(1.0f+expf(-x)); }
__device__ __forceinline__ float reluf(float x){ return x > 0.0f ? x : 0.0f; }

__device__ __forceinline__ float blkReduceSum(float v, float* red){
  int j = threadIdx.x;
  red[j] = v; __syncthreads();
  for (int s = (int)blockDim.x >> 1; s > 0; s >>= 1){
    if (j < s) red[j] += red[j+s];
    __syncthreads();
  }
  float r = red[0];
  __syncthreads();
  return r;
}

// ---------------------------------------------------------------------------
// WMMA f32 GEMM v2: C[g] = A[MxK] * W_g[KxHD] (+C if accumulate), row-major.
// 256 threads = 8 waves. Block stages A panel (16 x 64) and a transposed,
// padded B panel (128 x 64) in LDS; each wave owns one 16x16 C tile and
// issues 16 v_wmma_f32_16x16x4_f32 per K chunk from paired ds_load_b64 reads.
// ---------------------------------------------------------------------------
__global__ void __launch_bounds__(256)
gemm_wmma_k(const float* __restrict__ A,
            const float* __restrict__ Wg0, const float* __restrict__ Wg1,
            const float* __restrict__ Wg2, const float* __restrict__ Wg3,
            float* __restrict__ C, int M, int K, long long gateStride,
            int accumulate)
{
  int g = blockIdx.y;
  const float* W = (g==0)?Wg0:(g==1)?Wg1:(g==2)?Wg2:Wg3;
  float* Cg = C + (size_t)g * (size_t)gateStride;

  int m0   = blockIdx.x * 16;
  int tid  = threadIdx.x;
  int wv   = tid >> 5;      // wave id 0..7 -> N tile
  int lane = tid & 31;
  int hlf  = lane >> 4;     // lane half selects K pair (ISA 32-bit 16x4 layout)
  int lr   = lane & 15;
  int ncol = wv*16 + lr;

  __shared__ float As[16*LDP];    // A panel, padded
  __shared__ float Bt[HD*LDP];    // B panel transposed: Bt[col*LDP + k]

  v8f acc = {0.f,0.f,0.f,0.f,0.f,0.f,0.f,0.f};

  for (int kc = 0; kc < K; kc += KCHUNK){
    int kl = (K - kc < KCHUNK) ? (K - kc) : KCHUNK;
#if USE_ASYNC_LDS
    if (kl == KCHUNK){
      // A: 1024 elems, 4/thread ; Bt: 8192 elems, 32/thread (coalesced reads)
      for (int i = tid; i < 16*KCHUNK; i += 256){
        int r = i >> 6, cx = i & 63;
        async_ld_f32(&A[(size_t)(m0 + r)*K + kc + cx], &As[r*LDP + cx]);
      }
      for (int i = tid; i < HD*KCHUNK; i += 256){
        int kk = i >> 7, col = i & 127;
        async_ld_f32(&W[(size_t)(kc + kk)*HD + col], &Bt[col*LDP + kk]);
      }
      __builtin_amdgcn_s_wait_asynccnt(0);
    } else
#endif
    {
      for (int i = tid; i < 16*KCHUNK; i += 256){
        int r = i >> 6, cx = i & 63;
        As[r*LDP + cx] = (cx < kl) ? A[(size_t)(m0 + r)*K + kc + cx] : 0.0f;
      }
      for (int i = tid; i < HD*KCHUNK; i += 256){
        int kk = i >> 7, col = i & 127;
        Bt[col*LDP + kk] = (kk < kl) ? W[(size_t)(kc + kk)*HD + col] : 0.0f;
      }
    }
    __syncthreads();

    const float* ap = &As[lr*LDP   + 2*hlf];
    const float* bp = &Bt[ncol*LDP + 2*hlf];
#pragma unroll
    for (int k = 0; k < KCHUNK; k += 16){
      v2f a0,a1,a2,a3,b0,b1,b2,b3;
      a0.x = ap[k   ]; a0.y = ap[k+ 1]; b0.x = bp[k   ]; b0.y = bp[k+ 1];
      a1.x = ap[k+ 4]; a1.y = ap[k+ 5]; b1.x = bp[k+ 4]; b1.y = bp[k+ 5];
      a2.x = ap[k+ 8]; a2.y = ap[k+ 9]; b2.x = bp[k+ 8]; b2.y = bp[k+ 9];
      a3.x = ap[k+12]; a3.y = ap[k+13]; b3.x = bp[k+12]; b3.y = bp[k+13];
      acc = __builtin_amdgcn_wmma_f32_16x16x4_f32(false, a0, false, b0, (short)0, acc, false, false);
      acc = __builtin_amdgcn_wmma_f32_16x16x4_f32(false, a1, false, b1, (short)0, acc, false, false);
      acc = __builtin_amdgcn_wmma_f32_16x16x4_f32(false, a2, false, b2, (short)0, acc, false, false);
      acc = __builtin_amdgcn_wmma_f32_16x16x4_f32(false, a3, false, b3, (short)0, acc, false, false);
    }
    __syncthreads();
  }

  // D layout: lanes 0-15 rows M=r, lanes 16-31 rows M=r+8; col = ncol
  size_t base = (size_t)(m0 + 8*hlf)*HD + ncol;
  if (accumulate){
#pragma unroll
    for (int r = 0; r < 8; ++r) Cg[base + (size_t)r*HD] += acc[r];
  } else {
#pragma unroll
    for (int r = 0; r < 8; ++r) Cg[base + (size_t)r*HD] = acc[r];
  }
}

// ---------------------------------------------------------------------------
// Degree / normalization
// ---------------------------------------------------------------------------
__global__ void zero_k(float* __restrict__ p, int n){
  int i = blockIdx.x*blockDim.x + threadIdx.x;
  if (i < n) p[i] = 0.0f;
}
__global__ void deg_acc_k(const float* __restrict__ ew, const int* __restrict__ dst,
                          float* __restrict__ deg){
  int e = blockIdx.x*blockDim.x + threadIdx.x;
  if (e < ETOT) atomicAdd(&deg[dst[e]], ew[e]);
}
__global__ void dinv_k(float* __restrict__ d){
  int i = blockIdx.x*blockDim.x + threadIdx.x;
  if (i < NTOT) d[i] = rsqrtf(d[i] + 1.0f);   // deg includes self-loop weight 1
}

// agg[y] = dinv^2 * zw[y] + bias[y]   (y = side*4 + gate; x-side y<4)
__global__ void init_agg_k(const float* __restrict__ zwx, const float* __restrict__ zwh,
                           float* __restrict__ agg, const float* __restrict__ dinv,
                           Bias8 bias){
  int y = blockIdx.y;
  int i = blockIdx.x*blockDim.x + threadIdx.x;
  if (i >= NTOT*HD) return;
  int n = i / HD, h = i % HD;
  size_t NH_ = (size_t)NTOT*HD;
  const float* zw = (y < 4) ? (zwx + (size_t)y*NH_) : (zwh + (size_t)(y-4)*NH_);
  float d = dinv[n];
  float v = d*d*zw[i];
  const float* bp = bias.p[y];
  if (bp) v += bp[h];
  agg[(size_t)y*NH_ + i] = v;
}

// agg[y][dst] += dinv[src]*ew*dinv[dst] * zw[y][src]   (edge-parallel)
__global__ void scatter_k(const float* __restrict__ zwx, const float* __restrict__ zwh,
                          float* __restrict__ agg,
                          const int* __restrict__ src, const int* __restrict__ dst,
                          const float* __restrict__ ew, const float* __restrict__ dinv){
  int e = blockIdx.x, y = blockIdx.y, h = threadIdx.x;
  int s = src[e], d = dst[e];
  float nrm = dinv[s]*ew[e]*dinv[d];
  size_t NH_ = (size_t)NTOT*HD;
  const float* zw = (y < 4) ? (zwx + (size_t)y*NH_) : (zwh + (size_t)(y-4)*NH_);
  atomicAdd(&agg[(size_t)y*NH_ + (size_t)d*HD + h], nrm * zw[(size_t)s*HD + h]);
}

// Fused GCN-LSTM cell pointwise update (gate order: 0=i,1=f,2=o,3=m)
__global__ void combine_k(const float* __restrict__ agg, float* __restrict__ h,
                          float* __restrict__ c){
  int i = blockIdx.x*blockDim.x + threadIdx.x;
  if (i >= NTOT*HD) return;
  size_t S = (size_t)NTOT*HD;
  float gi = sigmf(reluf(agg[0*S+i]) + reluf(agg[4*S+i]));
  float gf = sigmf(reluf(agg[1*S+i]) + reluf(agg[5*S+i]));
  float go = sigmf(reluf(agg[2*S+i]) + reluf(agg[6*S+i]));
  float gm = tanhf(agg[3*S+i] + agg[7*S+i]);
  float cn = gf*c[i] + gi*gm;
  c[i] = cn;
  h[i] = go*tanhf(cn);
}

// ---------------------------------------------------------------------------
// DGPool
// ---------------------------------------------------------------------------
__global__ void __launch_bounds__(HD)
pool_score_k(const float* __restrict__ outp, const float* __restrict__ pv,
             float* __restrict__ sig){
  int b = blockIdx.x, j = threadIdx.x;
  __shared__ float red[HD];
  __shared__ float sv[NNODE];
  float v = pv[j];
  float ss = blkReduceSum(v*v, red);
  float inv = 1.0f/(sqrtf(ss) + 1e-8f);
  for (int n = j; n < NNODE; n += HD){
    const float* xr = outp + ((size_t)b*NNODE + n)*HD;
    float acc = 0.0f;
    for (int h = 0; h < HD; ++h) acc += xr[h]*pv[h];
    sv[n] = acc*inv;
  }
  __syncthreads();
  float pm = 0.0f;
  for (int n = j; n < NNODE; n += HD) pm += sv[n];
  float mean = blkReduceSum(pm, red) * (1.0f/NNODE);
  float pq = 0.0f;
  for (int n = j; n < NNODE; n += HD){ float d = sv[n]-mean; pq += d*d; }
  float sd = sqrtf(blkReduceSum(pq, red) * (1.0f/NNODE)) + 1e-8f;
  for (int n = j; n < NNODE; n += HD){
    float z = (sv[n]-mean)/sd;
    sig[(size_t)b*NNODE + n] = 1.0f/(1.0f+expf(-z));
  }
}

// Bitonic sort (descending) of 512 padded scores -> top-K indices + pool loss
__global__ void __launch_bounds__(512)
pool_sort_k(const float* __restrict__ sig, int* __restrict__ topidx,
            float* __restrict__ loss_b){
  int b = blockIdx.x, t = threadIdx.x;
  __shared__ float val[512];
  __shared__ short sid[512];
  __shared__ float red[512];
  val[t] = (t < NNODE) ? sig[(size_t)b*NNODE + t] : -1e30f;
  sid[t] = (short)t;
  __syncthreads();
  for (int ks = 2; ks <= 512; ks <<= 1){
    for (int jj = ks >> 1; jj > 0; jj >>= 1){
      int ixj = t ^ jj;
      if (ixj > t){
        bool desc = ((t & ks) == 0);
        float a = val[t], c = val[ixj];
        bool sw = desc ? (a < c) : (a > c);
        if (sw){ val[t]=c; val[ixj]=a; short q=sid[t]; sid[t]=sid[ixj]; sid[ixj]=q; }
      }
      __syncthreads();
    }
  }
  if (t < KTOP) topidx[b*KTOP + t] = (int)sid[t];
  float contrib = 0.0f;
  if (t < KTOP)       contrib = logf(val[t] + 1e-8f);
  else if (t < NNODE) contrib = logf(1.0f - val[t] + 1e-8f);
  red[t] = contrib; __syncthreads();
  for (int s = 256; s > 0; s >>= 1){ if (t < s) red[t] += red[t+s]; __syncthreads(); }
  if (t == 0) loss_b[b] = -red[0] * (1.0f/NNODE);
}

// pooled = mean over top-K of (x*sig); then layernorm -> hi
__global__ void __launch_bounds__(HD)
pool_gather_k(const float* __restrict__ outp, const float* __restrict__ sig,
              const int* __restrict__ topidx, const float* __restrict__ g,
              const float* __restrict__ bb, float* __restrict__ hi){
  int b = blockIdx.x, j = threadIdx.x;
  __shared__ float red[HD];
  float acc = 0.0f;
  for (int k = 0; k < KTOP; ++k){
    int n = topidx[b*KTOP + k];
    acc += outp[((size_t)b*NNODE + n)*HD + j] * sig[(size_t)b*NNODE + n];
  }
  acc *= (1.0f/KTOP);
  float mean = blkReduceSum(acc, red) * (1.0f/HD);
  float d = acc - mean;
  float var = blkReduceSum(d*d, red) * (1.0f/HD);
  hi[(size_t)b*HD + j] = d*rsqrtf(var + 1e-5f)*g[j] + bb[j];
}

// ---------------------------------------------------------------------------
// Time-series LSTM (persistent per-batch block; h in LDS, c in registers)
// ---------------------------------------------------------------------------
__global__ void __launch_bounds__(HD)
ts_lstm_k(const float* __restrict__ ts, const float* __restrict__ Wih,
          const float* __restrict__ Whh, const float* __restrict__ bih,
          const float* __restrict__ bhh, const float* __restrict__ lng,
          const float* __restrict__ lnb, const float* __restrict__ eg,
          const float* __restrict__ eb, float* __restrict__ lo)
{
  int b = blockIdx.x, j = threadIdx.x;
  __shared__ float xs[64];
  __shared__ float hs[HD];
  __shared__ float red[HD];
  float c = 0.0f;
  hs[j] = 0.0f;
  __syncthreads();
  for (int t = 0; t < TSL; ++t){
    const float* x = ts + ((size_t)b*TSL + t)*50;
    if (j < 50) xs[j] = x[j];
    __syncthreads();
    float m = 0.0f;
    for (int k = 0; k < 50; ++k) m += xs[k];
    m *= 0.02f;
    float v = 0.0f;
    for (int k = 0; k < 50; ++k){ float d = xs[k]-m; v += d*d; }
    v *= 0.02f;
    float rs = rsqrtf(v + 1e-5f);
    float gi = bih[j       ] + bhh[j       ];
    float gf = bih[j +   HD] + bhh[j +   HD];
    float gg = bih[j + 2*HD] + bhh[j + 2*HD];
    float go = bih[j + 3*HD] + bhh[j + 3*HD];
    for (int k = 0; k < 50; ++k){
      float xk = (xs[k]-m)*rs*lng[k] + lnb[k];
      gi += Wih[(size_t)(j       )*50 + k]*xk;
      gf += Wih[(size_t)(j +   HD)*50 + k]*xk;
      gg += Wih[(size_t)(j + 2*HD)*50 + k]*xk;
      go += Wih[(size_t)(j + 3*HD)*50 + k]*xk;
    }
    for (int k = 0; k < HD; ++k){
      float hk = hs[k];
      gi += Whh[(size_t)(j       )*HD + k]*hk;
      gf += Whh[(size_t)(j +   HD)*HD + k]*hk;
      gg += Whh[(size_t)(j + 2*HD)*HD + k]*hk;
      go += Whh[(size_t)(j + 3*HD)*HD + k]*hk;
    }
    c = sigmf(gf)*c + sigmf(gi)*tanhf(gg);
    float hn = sigmf(go)*tanhf(c);
    __syncthreads();
    hs[j] = hn;
    __syncthreads();
  }
  float hv = hs[j];
  float mean = blkReduceSum(hv, red) * (1.0f/HD);
  float d = hv - mean;
  float var = blkReduceSum(d*d, red) * (1.0f/HD);
  lo[(size_t)b*HD + j] = d*rsqrtf(var + 1e-5f)*eg[j] + eb[j];
}

// ---------------------------------------------------------------------------
// Final MLP classifier
// ---------------------------------------------------------------------------
__global__ void __launch_bounds__(HD)
mlp_k(const float* __restrict__ hi, const float* __restrict__ lo,
      const float* __restrict__ W1, const float* __restrict__ b1,
      const float* __restrict__ W2, const float* __restrict__ b2,
      const float* __restrict__ W3, const float* __restrict__ b3,
      const float* __restrict__ mg, const float* __restrict__ mb,
      float* __restrict__ preds)
{
  int b = blockIdx.x, j = threadIdx.x;
  __shared__ float fus[2*HD];
  __shared__ float m1[HD];
  __shared__ float red[HD];
  fus[j]      = hi[(size_t)b*HD + j];
  fus[j + HD] = lo[(size_t)b*HD + j];
  __syncthreads();
  float a = b1[j];
  for (int k = 0; k < 2*HD; ++k) a += fus[k]*W1[(size_t)k*HD + j];
  a = reluf(a);
  float mean = blkReduceSum(a, red) * (1.0f/HD);
  float d = a - mean;
  float var = blkReduceSum(d*d, red) * (1.0f/HD);
  m1[j] = d*rsqrtf(var + 1e-5f)*mg[j] + mb[j];
  __syncthreads();
  float m2 = 0.0f;
  if (j < 64){
    m2 = b2[j];
    for (int k = 0; k < HD; ++k) m2 += m1[k]*W2[(size_t)k*64 + j];
    m2 = reluf(m2);
  }
  red[j] = (j < 64) ? m2*W3[j] : 0.0f;
  __syncthreads();
  for (int s = 64; s > 0; s >>= 1){ if (j < s) red[j] += red[j+s]; __syncthreads(); }
  if (j == 0) preds[b] = red[0] + b3[0];
}

// Deterministic 16-way loss sum (no float atomics into d_out)
__global__ void finalize_loss_k(const float* __restrict__ loss_b, float* __restrict__ outl){
  if (threadIdx.x == 0 && blockIdx.x == 0){
    float s = 0.0f;
    for (int b = 0; b < BB; ++b) s += loss_b[b];
    *outl = s * (1.0f/BB);
  }
}

// ---------------------------------------------------------------------------
// Host orchestration
// ---------------------------------------------------------------------------
extern "C" void kernel_launch(void* const* d_in, const int* in_sizes, int n_in,
                              void* d_out, int out_size, void* d_ws, size_t ws_size,
                              hipStream_t stream)
{
  (void)in_sizes; (void)n_in; (void)out_size;
  // Inputs: setup_inputs() order; params dict flattened in insertion order.
  const float* x_seq  = (const float*)d_in[0];
  const float* ew_seq = (const float*)d_in[1];
  const float* h0     = (const float*)d_in[2];
  const float* c0     = (const float*)d_in[3];
  const float* ts     = (const float*)d_in[4];
  const int*   esrc   = (const int*)d_in[5];
  const int*   edst   = (const int*)d_in[6];
  const float* Wix = (const float*)d_in[7];
  const float* Wih = (const float*)d_in[8];
  const float* Wfx = (const float*)d_in[9];
  const float* Wfh = (const float*)d_in[10];
  const float* Wox = (const float*)d_in[11];
  const float* Woh = (const float*)d_in[12];
  const float* Wmx = (const float*)d_in[13];
  const float* Wmh = (const float*)d_in[14];
  const float* bix = (const float*)d_in[15];
  const float* bih = (const float*)d_in[16];
  const float* bfx = (const float*)d_in[17];
  const float* bfh = (const float*)d_in[18];
  const float* box_ = (const float*)d_in[19];
  const float* boh  = (const float*)d_in[20];
  const float* Mp1i0 = (const float*)d_in[21];
  const float* Mp2i0 = (const float*)d_in[22];
  const float* Mp2i1 = (const float*)d_in[23];
  const float* pool_v = (const float*)d_in[24];
  const float* gnn_g = (const float*)d_in[25];
  const float* gnn_b = (const float*)d_in[26];
  const float* lWih = (const float*)d_in[27];
  const float* lWhh = (const float*)d_in[28];
  const float* lbih = (const float*)d_in[29];
  const float* lbhh = (const float*)d_in[30];
  const float* in_g = (const float*)d_in[31];
  const float* in_b = (const float*)d_in[32];
  const float* emb_g = (const float*)d_in[33];
  const float* emb_b = (const float*)d_in[34];
  const float* W1 = (const float*)d_in[35];
  const float* b1 = (const float*)d_in[36];
  const float* W2 = (const float*)d_in[37];
  const float* b2 = (const float*)d_in[38];
  const float* W3 = (const float*)d_in[39];
  const float* b3 = (const float*)d_in[40];
  const float* mlp_g = (const float*)d_in[41];
  const float* mlp_b = (const float*)d_in[42];
  float* out = (float*)d_out;   // [0..15]=preds, [16]=pool_loss

  const size_t NH = (size_t)NTOT*HD;
  char* wp = (char*)d_ws;
  auto carve = [&](size_t nfloats)->float*{
    float* p = (float*)wp;
    wp += ((nfloats*sizeof(float)) + 255) & ~(size_t)255;
    return p;
  };
  float* zw    = carve(8*NH);   // [0..3]=x-side gates i,f,o,m ; [4..7]=h-side
  float* agg   = carve(8*NH);
  float* dinv  = carve(NTOT);
  float* h_cur = carve(NH);
  float* c_cur = carve(NH);
  float* h_p1  = carve(NH);
  float* h_p20 = carve(NH);
  float* h_p21 = carve(NH);
  float* outp  = carve(NH);
  float* sig   = carve((size_t)BB*NNODE);
  int*   topidx = (int*)carve((size_t)BB*KTOP);
  float* loss_b = carve(BB);
  float* hi    = carve((size_t)BB*HD);
  float* lo    = carve((size_t)BB*HD);
  // Optional XW precompute: x_t @ Wgx for all 8 timesteps x 4 gates (+105 MB).
  size_t used = (size_t)(wp - (char*)d_ws);
  bool precomp = (ws_size >= used + (size_t)TT*4*NH*sizeof(float) + 4096);
  float* xw = precomp ? carve((size_t)TT*4*NH) : nullptr;

  Bias8 bias;
  bias.p[0]=bix; bias.p[1]=bfx; bias.p[2]=box_; bias.p[3]=nullptr;
  bias.p[4]=bih; bias.p[5]=bfh; bias.p[6]=boh;  bias.p[7]=nullptr;

  dim3 blk256(256);
  int nhBlocks = (int)((NH + 255)/256);
  float* zwh = zw + 4*NH;

  if (precomp){
    for (int t = 0; t < TT; ++t){
      gemm_wmma_k<<<dim3(NTOT/16,4), blk256, 0, stream>>>(
          x_seq + (size_t)t*NTOT*FDIM, Wix, Wfx, Wox, Wmx,
          xw + (size_t)t*4*NH, NTOT, FDIM, (long long)NH, 0);
    }
  }

  auto run_stream = [&](int t0, int stride, int nsteps, float* hOut){
    (void)hipMemcpyAsync(h_cur, h0, NH*sizeof(float), hipMemcpyDeviceToDevice, stream);
    (void)hipMemcpyAsync(c_cur, c0, NH*sizeof(float), hipMemcpyDeviceToDevice, stream);
    for (int s = 0; s < nsteps; ++s){
      int t = t0 + s*stride;
      const float* xt  = x_seq  + (size_t)t*NTOT*FDIM;
      const float* ewt = ew_seq + (size_t)t*ETOT;
      zero_k   <<<(NTOT+255)/256, blk256, 0, stream>>>(dinv, NTOT);
      deg_acc_k<<<(ETOT+255)/256, blk256, 0, stream>>>(ewt, edst, dinv);
      dinv_k   <<<(NTOT+255)/256, blk256, 0, stream>>>(dinv);
      const float* zwx;
      if (precomp){
        zwx = xw + (size_t)t*4*NH;
      } else {
        gemm_wmma_k<<<dim3(NTOT/16,4), blk256, 0, stream>>>(
            xt, Wix, Wfx, Wox, Wmx, zw, NTOT, FDIM, (long long)NH, 0);
        zwx = zw;
      }
      gemm_wmma_k<<<dim3(NTOT/16,4), blk256, 0, stream>>>(
          h_cur, Wih, Wfh, Woh, Wmh, zwh, NTOT, HD, (long long)NH, 0);
      init_agg_k<<<dim3(nhBlocks,8), blk256, 0, stream>>>(zwx, zwh, agg, dinv, bias);
      scatter_k <<<dim3(ETOT,8), dim3(HD), 0, stream>>>(zwx, zwh, agg, esrc, edst, ewt, dinv);
      combine_k <<<nhBlocks, blk256, 0, stream>>>(agg, h_cur, c_cur);
    }
    (void)hipMemcpyAsync(hOut, h_cur, NH*sizeof(float), hipMemcpyDeviceToDevice, stream);
  };

  run_stream(0, 1, TT,   h_p1);
  run_stream(0, 2, TT/2, h_p20);
  run_stream(1, 2, TT/2, h_p21);

  // output = h_p1@Mp1i0 + h_p20@Mp2i0 + h_p21@Mp2i1
  gemm_wmma_k<<<dim3(NTOT/16,1), blk256, 0, stream>>>(
      h_p1,  Mp1i0, Mp1i0, Mp1i0, Mp1i0, outp, NTOT, HD, 0, 0);
  gemm_wmma_k<<<dim3(NTOT/16,1), blk256, 0, stream>>>(
      h_p20, Mp2i0, Mp2i0, Mp2i0, Mp2i0, outp, NTOT, HD, 0, 1);
  gemm_wmma_k<<<dim3(NTOT/16,1), blk256, 0, stream>>>(
      h_p21, Mp2i1, Mp2i1, Mp2i1, Mp2i1, outp, NTOT, HD, 0, 1);

  pool_score_k <<<BB, dim3(HD),  0, stream>>>(outp, pool_v, sig);
  pool_sort_k  <<<BB, dim3(512), 0, stream>>>(sig, topidx, loss_b);
  pool_gather_k<<<BB, dim3(HD),  0, stream>>>(outp, sig, topidx, gnn_g, gnn_b, hi);
  ts_lstm_k    <<<BB, dim3(HD),  0, stream>>>(ts, lWih, lWhh, lbih, lbhh,
                                              in_g, in_b, emb_g, emb_b, lo);
  mlp_k        <<<BB, dim3(HD),  0, stream>>>(hi, lo, W1, b1, W2, b2, W3, b3,
                                              mlp_g, mlp_b, out);
  finalize_loss_k<<<1, 32, 0, stream>>>(loss_b, out + BB);
}